// ElementRelationships_18098992185823
// MI455X (gfx1250) — compile-verified
//
#include <hip/hip_runtime.h>
#include <hip/hip_bf16.h>
#include <stdint.h>

// out[b,t,n,f] = in[b,t,n,f] * (ALPHA+BETA) * (n < set_size[b,t])
// B=32, T=64, N=128, F=256  (N, F powers of two -> shift/mask index math)
//
// Pure streaming op: 512 MiB total HBM traffic, ~22us floor at 23.3 TB/s.
// Strategy: CDNA5 async-to-LDS B128 loads (ASYNCcnt-tracked, double-buffered,
// non-temporal) + non-temporal B128 stores + wave-uniform scalar mask fetch.
// No WMMA: AI = 0.125 FLOP/byte; the matrix pipes are irrelevant here.

typedef float f32x4 __attribute__((ext_vector_type(4)));

#define TPB    256
#define BLOCKS 4096

__global__ __launch_bounds__(TPB) void ElementRelationships_mask_scale_async(
    const float* __restrict__ in,
    const long long* __restrict__ set_size,   // int64 per reference
    float* __restrict__ out,
    int iters,
    long long stride_vec4)                    // grid stride, in float4 units
{
    __shared__ f32x4 buf[2][TPB];             // 8 KiB double buffer

    const int tid = threadIdx.x;
    long long idx = (long long)blockIdx.x * TPB + tid;   // float4 index

    // Wave-relative LDS byte offsets (generic-pointer low 32 bits == LDS offset)
    const unsigned lds0 = (unsigned)(uintptr_t)&buf[0][tid];
    const unsigned lds1 = (unsigned)(uintptr_t)&buf[1][tid];

    // Stage iteration 0 into buffer 0 (read-once data -> non-temporal).
    if (iters > 0) {
        unsigned long long g = (unsigned long long)(uintptr_t)(in + idx * 4);
        asm volatile("global_load_async_to_lds_b128 %0, %1, off th:TH_LOAD_NT"
                     :: "v"(lds0), "v"(g) : "memory");
    }

    for (int it = 0; it < iters; ++it) {
        long long nidx = idx + stride_vec4;
        if (it + 1 < iters) {
            // Stage next tile into the other buffer, then wait until only
            // that one is outstanding -> current tile is complete in LDS.
            unsigned ldsn = ((it + 1) & 1) ? lds1 : lds0;
            unsigned long long g =
                (unsigned long long)(uintptr_t)(in + nidx * 4);
            asm volatile("global_load_async_to_lds_b128 %0, %1, off th:TH_LOAD_NT"
                         :: "v"(ldsn), "v"(g) : "memory");
            asm volatile("s_wait_asynccnt 0x1" ::: "memory");
        } else {
            asm volatile("s_wait_asynccnt 0x0" ::: "memory");
        }

        f32x4 v = buf[it & 1][tid];           // ds_load_b128 (own data, no barrier)

        // idx is in float4 units; F/4 = 64 vec4 per row, so 64 consecutive
        // lanes share a row: row/n/bt/s are wave-uniform. Force them scalar
        // so set_size comes in via one s_load_b64 instead of a 32-lane gather.
        // (row fits in 32 bits for any tensor < 2^31 rows.)
        int row  = __builtin_amdgcn_readfirstlane((int)(idx >> 6));
        int n    = row & 127;                 // N = 128
        int bt   = row >> 7;
        long long ssv = set_size[bt];         // uniform -> scalar load
        float s = ((long long)n < ssv) ? 1.1f : 0.0f;   // ALPHA+BETA
        v *= s;

        __builtin_nontemporal_store(v, (f32x4*)(out + idx * 4));
        idx = nidx;
    }
}

// Scalar tail for any elements not covered by the vectorized main kernel
// (not launched for the reference shapes, which divide exactly).
__global__ void ElementRelationships_mask_scale_tail(
    const float* __restrict__ in,
    const long long* __restrict__ set_size,
    float* __restrict__ out,
    long long start, long long total)
{
    long long i = start + (long long)blockIdx.x * blockDim.x + threadIdx.x;
    if (i < total) {
        long long row = i >> 8;               // F = 256
        int       n   = (int)(row & 127);     // N = 128
        long long bt  = row >> 7;
        float s = ((long long)n < set_size[bt]) ? 1.1f : 0.0f;
        out[i] = in[i] * s;
    }
}

extern "C" void kernel_launch(void* const* d_in, const int* in_sizes, int n_in,
                              void* d_out, int out_size, void* d_ws, size_t ws_size,
                              hipStream_t stream)
{
    const float*     in  = (const float*)d_in[0];
    const long long* ss  = (const long long*)d_in[1];   // int64 per reference
    float*           out = (float*)d_out;

    const long long total = (long long)in_sizes[0];     // B*T*N*F
    const long long nvec4 = total >> 2;                 // F=256 -> divisible by 4
    const long long per   = (long long)TPB * BLOCKS;    // vec4 per grid iteration
    const int iters = (int)(nvec4 / per);               // = 16 for reference shapes

    if (iters > 0) {
        ElementRelationships_mask_scale_async<<<BLOCKS, TPB, 0, stream>>>(
            in, ss, out, iters, per);
    }

    const long long done = (long long)iters * per * 4;  // elements covered
    const long long rem  = total - done;
    if (rem > 0) {
        int tb = (int)((rem + 255) / 256);
        ElementRelationships_mask_scale_tail<<<tb, 256, 0, stream>>>(
            in, ss, out, done, total);
    }
}